// TransformerLayer_35708358099442
// MI455X (gfx1250) — compile-verified
//
#include <hip/hip_runtime.h>

// ---------------------------------------------------------------------------
// Transformer layer for MI455X (gfx1250, wave32).
// All GEMMs via V_WMMA_F32_16X16X4_F32 (fp32-exact vs reference, and the
// workload is gather-bandwidth-bound anyway: ~1.6GB random K/V gather at
// 23.3 TB/s dominates the ~41 GFLOP of matmul).
// ---------------------------------------------------------------------------

typedef float v8f __attribute__((ext_vector_type(8)));
typedef float v2f __attribute__((ext_vector_type(2)));

#define N_TOK 100000
#define DM    128
#define KS    16
#define HD    32
#define DFF   512
#define QKVD  384

static __device__ __forceinline__ v8f wmma_f32(v2f a, v2f b, v8f c) {
  // (neg_a, A, neg_b, B, c_mod, C, reuse_a, reuse_b) — same pattern as the
  // probed 8-arg f16/bf16 builtins. A,B = 16x4 / 4x16 f32 = 2 VGPRs per lane.
  return __builtin_amdgcn_wmma_f32_16x16x4_f32(false, a, false, b, (short)0, c,
                                               false, false);
}

static __device__ __forceinline__ v2f ld2(const float* p) {
  return *reinterpret_cast<const v2f*>(p);
}

// ---------------- Kernel 1: QKV projection (x @ in_proj_w.T + b) -----------
// One wave -> one 16x16 output tile of the (N, 384) interleaved QKV tensor.
__global__ void qkv_kernel(const float* __restrict__ x,
                           const float* __restrict__ w,
                           const float* __restrict__ bias,
                           float* __restrict__ qkv) {
  const int lane = threadIdx.x;
  const int m = lane & 15, half = lane >> 4;
  const int row0 = blockIdx.x * 16;
  const int col0 = blockIdx.y * 16;
  v8f acc = {};
  const float* xr = x + (size_t)(row0 + m) * DM;
  const float* wr = w + (size_t)(col0 + m) * DM;
#pragma unroll
  for (int kk = 0; kk < DM / 4; ++kk) {
    const int kb = kk * 4 + half * 2;
    acc = wmma_f32(ld2(xr + kb), ld2(wr + kb), acc);
  }
  const float bv = bias[col0 + m];
#pragma unroll
  for (int r = 0; r < 8; ++r) {
    const int row = row0 + r + 8 * half;
    qkv[(size_t)row * QKVD + col0 + m] = acc[r] + bv;
  }
}

// ---------------- Kernel 2: sampled attention ------------------------------
// One wave per token. 64 scores -> 2 per lane (heads interleave with lane/16),
// 16-lane shuffle softmax, weights through LDS, coalesced V accumulation.
__global__ void attn_kernel(const float* __restrict__ qkv,
                            const int* __restrict__ samp,
                            float* __restrict__ aout) {
  __shared__ float qs[8][DM];
  __shared__ float wsm[8][64];
  __shared__ int   ism[8][KS];
  const int lane = threadIdx.x & 31;
  const int wid  = threadIdx.x >> 5;
  const int tok  = blockIdx.x * 8 + wid;

  const float* qrow = qkv + (size_t)tok * QKVD;
#pragma unroll
  for (int t = 0; t < 4; ++t) qs[wid][lane + 32 * t] = qrow[lane + 32 * t];

  const int g = lane >> 4;     // lane half -> head group
  const int j = lane & 15;     // sampled-key index
  const int idx = samp[tok * KS + j];
  if (lane < KS) ism[wid][lane] = idx;
  const int h0 = g, h1 = 2 + g;

  const float* krow = qkv + (size_t)idx * QKVD + DM;
  float s0 = 0.f, s1 = 0.f;
#pragma unroll
  for (int d = 0; d < HD; ++d) {
    s0 += qs[wid][h0 * HD + d] * krow[h0 * HD + d];
    s1 += qs[wid][h1 * HD + d] * krow[h1 * HD + d];
  }
  const float scale = 0.17677669529663687f;  // 1/sqrt(32)
  s0 *= scale; s1 *= scale;

  float m0 = s0, m1 = s1;
#pragma unroll
  for (int msk = 8; msk >= 1; msk >>= 1) {
    m0 = fmaxf(m0, __shfl_xor(m0, msk, 32));
    m1 = fmaxf(m1, __shfl_xor(m1, msk, 32));
  }
  const float e0 = __expf(s0 - m0), e1 = __expf(s1 - m1);
  float z0 = e0, z1 = e1;
#pragma unroll
  for (int msk = 8; msk >= 1; msk >>= 1) {
    z0 += __shfl_xor(z0, msk, 32);
    z1 += __shfl_xor(z1, msk, 32);
  }
  wsm[wid][h0 * KS + j] = e0 / z0;
  wsm[wid][h1 * KS + j] = e1 / z1;
  __syncthreads();

  float acc[4] = {0.f, 0.f, 0.f, 0.f};
#pragma unroll
  for (int jj = 0; jj < KS; ++jj) {
    const float* vrow = qkv + (size_t)ism[wid][jj] * QKVD + 2 * DM;
#pragma unroll
    for (int t = 0; t < 4; ++t) {
      const int e = lane + 32 * t;               // output element, head = e>>5
      acc[t] += wsm[wid][(e >> 5) * KS + jj] * vrow[e];
    }
  }
#pragma unroll
  for (int t = 0; t < 4; ++t)
    aout[(size_t)tok * DM + lane + 32 * t] = acc[t];
}

// ---------------- Kernel 3: out-proj + bias + residual ---------------------
__global__ void oproj_kernel(const float* __restrict__ a,
                             const float* __restrict__ w,
                             const float* __restrict__ bias,
                             const float* __restrict__ x,
                             float* __restrict__ y) {
  const int lane = threadIdx.x;
  const int m = lane & 15, half = lane >> 4;
  const int row0 = blockIdx.x * 16;
  const int col0 = blockIdx.y * 16;
  v8f acc = {};
  const float* ar = a + (size_t)(row0 + m) * DM;
  const float* wr = w + (size_t)(col0 + m) * DM;
#pragma unroll
  for (int kk = 0; kk < DM / 4; ++kk) {
    const int kb = kk * 4 + half * 2;
    acc = wmma_f32(ld2(ar + kb), ld2(wr + kb), acc);
  }
  const float bv = bias[col0 + m];
#pragma unroll
  for (int r = 0; r < 8; ++r) {
    const size_t o = (size_t)(row0 + r + 8 * half) * DM + col0 + m;
    y[o] = acc[r] + bv + x[o];
  }
}

// ---------------- Kernel 4/6: layernorm (wave per row) ---------------------
__global__ void ln_kernel(const float* __restrict__ in,
                          const float* __restrict__ g,
                          const float* __restrict__ b,
                          float* __restrict__ out) {
  const int lane = threadIdx.x & 31;
  const int wid  = threadIdx.x >> 5;
  const size_t row = (size_t)blockIdx.x * 8 + wid;
  const float* p = in + row * DM;
  float v[4];
#pragma unroll
  for (int t = 0; t < 4; ++t) v[t] = p[lane + 32 * t];
  float s = v[0] + v[1] + v[2] + v[3];
#pragma unroll
  for (int msk = 16; msk >= 1; msk >>= 1) s += __shfl_xor(s, msk, 32);
  const float mu = s * (1.0f / DM);
  float q = 0.f;
#pragma unroll
  for (int t = 0; t < 4; ++t) { const float d = v[t] - mu; q += d * d; }
#pragma unroll
  for (int msk = 16; msk >= 1; msk >>= 1) q += __shfl_xor(q, msk, 32);
  const float rs = rsqrtf(q * (1.0f / DM) + 1e-5f);
  float* po = out + row * DM;
#pragma unroll
  for (int t = 0; t < 4; ++t) {
    const int c = lane + 32 * t;
    po[c] = (v[t] - mu) * rs * g[c] + b[c];
  }
}

// ---------------- Kernel 5: fused FFN (x1@W1.T -> relu -> @W2.T + res) -----
// One wave per 16-row tile. h-tiles (16x16) are re-striped C->A layout via
// LDS and immediately consumed by the second GEMM's 8 persistent accumulators,
// so the (N,512) hidden tensor never touches global memory.
__global__ void ffn_kernel(const float* __restrict__ x1,
                           const float* __restrict__ w1,
                           const float* __restrict__ b1,
                           const float* __restrict__ w2,
                           const float* __restrict__ b2,
                           float* __restrict__ y) {
  __shared__ float xs[16 * DM];   // x1 tile, row-major
  __shared__ float hs[16 * 16];   // one relu'd hidden tile
  const int lane = threadIdx.x;
  const int m = lane & 15, half = lane >> 4;
  const int row0 = blockIdx.x * 16;

  for (int i = 0; i < 64; ++i)
    xs[i * 32 + lane] = x1[(size_t)row0 * DM + i * 32 + lane];

  const v8f vz = {};
  v8f oacc[8];
#pragma unroll
  for (int c = 0; c < 8; ++c) oacc[c] = vz;

  for (int t = 0; t < DFF / 16; ++t) {
    v8f hacc = {};
    const float* wr = w1 + (size_t)(t * 16 + m) * DM;
#pragma unroll
    for (int kk = 0; kk < DM / 4; ++kk) {
      const int kb = kk * 4 + half * 2;
      hacc = wmma_f32(ld2(&xs[m * DM + kb]), ld2(wr + kb), hacc);
    }
    const float bv = b1[t * 16 + m];
#pragma unroll
    for (int r = 0; r < 8; ++r)
      hs[(r + 8 * half) * 16 + m] = fmaxf(hacc[r] + bv, 0.f);
    // same-wave LDS RAW/WAR: compiler orders via DScnt waits
#pragma unroll
    for (int c = 0; c < 8; ++c) {
      const float* w2r = w2 + (size_t)(c * 16 + m) * DFF + t * 16;
#pragma unroll
      for (int k2 = 0; k2 < 4; ++k2) {
        const int kb2 = k2 * 4 + half * 2;
        oacc[c] = wmma_f32(ld2(&hs[m * 16 + kb2]), ld2(w2r + kb2), oacc[c]);
      }
    }
  }
#pragma unroll
  for (int c = 0; c < 8; ++c) {
    const float bv = b2[c * 16 + m];
#pragma unroll
    for (int r = 0; r < 8; ++r) {
      const size_t o = (size_t)(row0 + r + 8 * half) * DM + c * 16 + m;
      y[o] = oacc[c][r] + bv + x1[o];
    }
  }
}

// ---------------------------------------------------------------------------
extern "C" void kernel_launch(void* const* d_in, const int* in_sizes, int n_in,
                              void* d_out, int out_size, void* d_ws,
                              size_t ws_size, hipStream_t stream) {
  const float* x   = (const float*)d_in[0];
  const int*   smp = (const int*)  d_in[1];
  const float* inw = (const float*)d_in[2];
  const float* inb = (const float*)d_in[3];
  const float* ow  = (const float*)d_in[4];
  const float* ob  = (const float*)d_in[5];
  const float* w1  = (const float*)d_in[6];
  const float* b1  = (const float*)d_in[7];
  const float* w2  = (const float*)d_in[8];
  const float* b2  = (const float*)d_in[9];
  const float* g1  = (const float*)d_in[10];
  const float* lb1 = (const float*)d_in[11];
  const float* g2  = (const float*)d_in[12];
  const float* lb2 = (const float*)d_in[13];
  float* out = (float*)d_out;
  float* ws  = (float*)d_ws;

  const size_t ND = (size_t)N_TOK * DM;
  float* qkv  = ws;            // [0, 3ND)   interleaved (N,384)
  float* aout = ws + 3 * ND;   // [3ND, 4ND)
  float* y1   = ws;            // reuses qkv region (dead after attn)
  float* x1   = ws + ND;
  float* y2   = ws + 2 * ND;
  (void)in_sizes; (void)n_in; (void)out_size; (void)ws_size;

  qkv_kernel <<<dim3(N_TOK / 16, QKVD / 16), dim3(32),  0, stream>>>(x, inw, inb, qkv);
  attn_kernel<<<dim3(N_TOK / 8),             dim3(256), 0, stream>>>(qkv, smp, aout);
  oproj_kernel<<<dim3(N_TOK / 16, DM / 16),  dim3(32),  0, stream>>>(aout, ow, ob, x, y1);
  ln_kernel  <<<dim3(N_TOK / 8),             dim3(256), 0, stream>>>(y1, g1, lb1, x1);
  ffn_kernel <<<dim3(N_TOK / 16),            dim3(32),  0, stream>>>(x1, w1, b1, w2, b2, y2);
  ln_kernel  <<<dim3(N_TOK / 8),             dim3(256), 0, stream>>>(y2, g2, lb2, out);
}